// RPQLinear_3255585210641
// MI455X (gfx1250) — compile-verified
//
#include <hip/hip_runtime.h>

typedef float v2f __attribute__((ext_vector_type(2)));
typedef float v4f __attribute__((ext_vector_type(4)));
typedef float v8f __attribute__((ext_vector_type(8)));

// Problem constants (from reference)
#define H_CB   8
#define CODES  256
#define DCB    512
#define KDIM   4096   // IN
#define NDIM   4096   // OUT = H*D
#define MDIM   8192   // B*S

// Tiling
#define BM  128
#define BN  128
#define BK  32
#define PAD 4
#define LDT (BK + PAD)   // 36 floats: 144B row stride -> 16B aligned, conflict-free

__global__ __launch_bounds__(256) void rpq_gemm_wmma(
    const float* __restrict__ x,
    const int*   __restrict__ indices,
    const float* __restrict__ codebooks,
    const float* __restrict__ bias,
    float*       __restrict__ out)
{
    __shared__ float xs[BM * LDT];   // x tile, row-major [m][k]
    __shared__ float bs[BN * LDT];   // gathered weight tile, transposed [n][k]

    const int tid  = threadIdx.x;
    const int lane = tid & 31;
    const int wave = tid >> 5;
    const int hi   = lane >> 4;   // lane half (0/1)
    const int lr   = lane & 15;

    const int mbase = blockIdx.y * BM;
    const int nbase = blockIdx.x * BN;
    const int h     = nbase >> 9;     // which codebook (D=512 per codebook)
    const int dbase = nbase & 511;    // offset inside codebook row

    const int wm = wave & 1;    // 2 wave-rows -> 64 m each
    const int wn = wave >> 1;   // 4 wave-cols -> 32 n each

    // per-thread staging assignments
    const int xc4  = tid & 7;     // x tile: 8 float4 per row
    const int xrow = tid >> 3;    // +32 per pass
    const int bk   = tid >> 3;    // B tile row (k), 0..31
    const int bc8  = tid & 7;     // B tile: float4 column group base

    v8f acc[4][2];
    const v8f vzero = {};
#pragma unroll
    for (int i = 0; i < 4; ++i)
#pragma unroll
        for (int j = 0; j < 2; ++j)
            acc[i][j] = vzero;

    // register double-buffer for the incoming tile
    v4f xv[4], bv[4];

    // ---- prefetch tile 0 into registers
    {
        int row = xrow;
#pragma unroll
        for (int p = 0; p < 4; ++p) {
            xv[p] = *(const v4f*)(x + (size_t)(mbase + row) * KDIM + 0 + xc4 * 4);
            row += 32;
        }
        const int idx = indices[h * KDIM + 0 + bk];
        const float* src = codebooks + ((size_t)(h * CODES + idx)) * DCB + dbase;
#pragma unroll
        for (int p = 0; p < 4; ++p)
            bv[p] = *(const v4f*)(src + (bc8 + 8 * p) * 4);
    }

    for (int kk = 0; kk < KDIM; kk += BK) {
        // ---- WAR: previous compute must be done reading LDS
        __syncthreads();

        // ---- dump register buffer into LDS
        {
            int row = xrow;
#pragma unroll
            for (int p = 0; p < 4; ++p) {
                *(v4f*)(&xs[row * LDT + xc4 * 4]) = xv[p];
                row += 32;
            }
#pragma unroll
            for (int p = 0; p < 4; ++p) {
                const int c4 = bc8 + 8 * p;          // cols c4*4 .. c4*4+3
#pragma unroll
                for (int j = 0; j < 4; ++j)
                    bs[(c4 * 4 + j) * LDT + bk] = bv[p][j];
            }
        }
        __syncthreads();

        // ---- prefetch next tile while this chunk computes (hides HBM/L2 latency)
        if (kk + BK < KDIM) {
            const int kn = kk + BK;
            int row = xrow;
#pragma unroll
            for (int p = 0; p < 4; ++p) {
                xv[p] = *(const v4f*)(x + (size_t)(mbase + row) * KDIM + kn + xc4 * 4);
                row += 32;
            }
            const int idx = indices[h * KDIM + kn + bk];
            const float* src = codebooks + ((size_t)(h * CODES + idx)) * DCB + dbase;
#pragma unroll
            for (int p = 0; p < 4; ++p)
                bv[p] = *(const v4f*)(src + (bc8 + 8 * p) * 4);
        }

        // ---- compute: 8 k-steps of v_wmma_f32_16x16x4_f32 per 16x16 tile
#pragma unroll
        for (int ks = 0; ks < BK; ks += 4) {
            const int kidx = ks + hi * 2;   // VGPR0 holds k=hi*2, VGPR1 holds k=hi*2+1
            v2f a[4], b[2];
#pragma unroll
            for (int i = 0; i < 4; ++i)
                a[i] = *(const v2f*)(&xs[(wm * 64 + i * 16 + lr) * LDT + kidx]);
#pragma unroll
            for (int j = 0; j < 2; ++j)
                b[j] = *(const v2f*)(&bs[(wn * 32 + j * 16 + lr) * LDT + kidx]);
#pragma unroll
            for (int i = 0; i < 4; ++i)
#pragma unroll
                for (int j = 0; j < 2; ++j)
                    acc[i][j] = __builtin_amdgcn_wmma_f32_16x16x4_f32(
                        false, a[i], false, b[j], (short)0, acc[i][j], false, false);
        }
    }

    // ---- epilogue: bias add + store (C/D layout: m = v + 8*half, n = lr)
#pragma unroll
    for (int j = 0; j < 2; ++j) {
        const int   n  = nbase + wn * 32 + j * 16 + lr;
        const float bv2 = bias[n];
#pragma unroll
        for (int i = 0; i < 4; ++i) {
            const int mrow = mbase + wm * 64 + i * 16 + hi * 8;
#pragma unroll
            for (int v = 0; v < 8; ++v)
                out[(size_t)(mrow + v) * NDIM + n] = acc[i][j][v] + bv2;
        }
    }
}

extern "C" void kernel_launch(void* const* d_in, const int* in_sizes, int n_in,
                              void* d_out, int out_size, void* d_ws, size_t ws_size,
                              hipStream_t stream) {
    const float* x         = (const float*)d_in[0];
    const int*   indices   = (const int*)  d_in[1];
    const float* codebooks = (const float*)d_in[2];
    const float* bias      = (const float*)d_in[3];
    float*       out       = (float*)d_out;

    dim3 grid(NDIM / BN, MDIM / BM);   // (32, 64)
    rpq_gemm_wmma<<<grid, 256, 0, stream>>>(x, indices, codebooks, bias, out);
}